// AdaptiveKANLayer_42777874268725
// MI455X (gfx1250) — compile-verified
//
#include <hip/hip_runtime.h>
#include <hip/hip_bf16.h>

typedef float v2f __attribute__((ext_vector_type(2)));
typedef float v8f __attribute__((ext_vector_type(8)));

#define BATCH   4096
#define IN_DIM  256
#define OUT_DIM 256
#define NKNOTS  9
#define NBASIS  5
#define KPAD    8   // pad K=5 -> 8 for two 16x16x4 f32 WMMAs

// ---------------------------------------------------------------------------
// Kernel 1: Cox-de-Boor cubic B-spline basis, padded to 8 along n.
// basisPad[(b*IN_DIM + i)*KPAD + n], n=5..7 zero.
// ---------------------------------------------------------------------------
__global__ __launch_bounds__(256) void kan_basis_kernel(
    const float* __restrict__ x, const float* __restrict__ grids,
    float* __restrict__ basisPad) {
  int idx = blockIdx.x * blockDim.x + threadIdx.x;   // b*IN_DIM + i
  int i = idx & (IN_DIM - 1);
  int b = idx >> 8;
  float xx = x[(size_t)b * IN_DIM + i];

  float t[NKNOTS];
#pragma unroll
  for (int k = 0; k < NKNOTS; ++k) t[k] = grids[(size_t)i * NKNOTS + k];

  float bas[NKNOTS - 1];
#pragma unroll
  for (int j = 0; j < NKNOTS - 1; ++j)
    bas[j] = (xx >= t[j] && xx < t[j + 1]) ? 1.0f : 0.0f;

#pragma unroll
  for (int d = 1; d <= 3; ++d) {
    int nb = NKNOTS - 1 - d;
#pragma unroll
    for (int j = 0; j < NKNOTS - 2; ++j) {   // in-place, increasing j is safe
      if (j < nb) {
        float left  = (xx - t[j]) / (t[j + d] - t[j]);
        float right = (t[j + d + 1] - xx) / (t[j + d + 1] - t[j + 1]);
        bas[j] = left * bas[j] + right * bas[j + 1];
      }
    }
  }

  float* o = basisPad + (size_t)idx * KPAD;
  o[0] = bas[0]; o[1] = bas[1]; o[2] = bas[2]; o[3] = bas[3]; o[4] = bas[4];
  o[5] = 0.0f;   o[6] = 0.0f;   o[7] = 0.0f;
}

// ---------------------------------------------------------------------------
// Kernel 2: repack spline_coeffs (i,o,5) -> 8-padded (i,o,8) for aligned
// float2 WMMA-B fragment loads.
// ---------------------------------------------------------------------------
__global__ __launch_bounds__(256) void kan_repack_kernel(
    const float* __restrict__ coeffs, float* __restrict__ coeffPad) {
  int idx = blockIdx.x * blockDim.x + threadIdx.x;   // i*OUT_DIM + o
  const float* s = coeffs + (size_t)idx * NBASIS;
  float* d = coeffPad + (size_t)idx * KPAD;
  d[0] = s[0]; d[1] = s[1]; d[2] = s[2]; d[3] = s[3]; d[4] = s[4];
  d[5] = 0.0f; d[6] = 0.0f; d[7] = 0.0f;
}

// ---------------------------------------------------------------------------
// Kernel 3: acts[b,i,o] = basis[b,i,:] . C[i,o,:]  via V_WMMA_F32_16X16X4_F32
// (K padded 5->8, two chained WMMAs), y[b,o] = sum_i acts[b,i,o].
// One wave per (16-batch b-tile, 16-output o-tile); 8 waves per block cover
// 8 o-tiles. grid = (OUT_DIM/128, BATCH/16).
// ---------------------------------------------------------------------------
__global__ __launch_bounds__(256) void kan_acts_wmma_kernel(
    const float* __restrict__ basisPad, const float* __restrict__ coeffPad,
    float* __restrict__ y, float* __restrict__ acts) {
  const int lane  = threadIdx.x & 31;
  const int wave  = threadIdx.x >> 5;
  const int otile = blockIdx.x * 8 + wave;        // 0..15
  const int btile = blockIdx.y;                   // 0..255
  const int bbase = btile * 16;
  const int obase = otile * 16;

  const int nlo    = lane & 15;                   // A row / B,C col within tile
  const int k0     = (lane < 16) ? 0 : 2;         // K split for A/B fragments
  const int rowOff = (lane < 16) ? 0 : 8;         // C/D row offset

  // A fragment base: basisPad[((bbase+nlo)*IN_DIM + i)*KPAD + k0]
  const float* aBase = basisPad + (size_t)(bbase + nlo) * IN_DIM * KPAD + k0;
  // B fragment base: coeffPad[(i*OUT_DIM + obase + nlo)*KPAD + k0]
  const float* bBase = coeffPad + (size_t)(obase + nlo) * KPAD + k0;
  // acts per-lane store base (+= i*OUT_DIM, += r*IN_DIM*OUT_DIM inside loop)
  float* actBase = acts + (size_t)(bbase + rowOff) * IN_DIM * OUT_DIM
                        + (size_t)obase + nlo;

  v8f acc = {};
  for (int i = 0; i < IN_DIM; ++i) {
    v2f a1 = *(const v2f*)(aBase + (size_t)i * KPAD);
    v2f a2 = *(const v2f*)(aBase + (size_t)i * KPAD + 4);
    v2f b1 = *(const v2f*)(bBase + (size_t)i * OUT_DIM * KPAD);
    v2f b2 = *(const v2f*)(bBase + (size_t)i * OUT_DIM * KPAD + 4);

    v8f z = {};
    // D = A2*B2 + 0   (K = 4..7, only K=4 nonzero)
    v8f tacc = __builtin_amdgcn_wmma_f32_16x16x4_f32(
        false, a2, false, b2, (short)0, z, false, false);
    // D = A1*B1 + D   (K = 0..3)
    v8f d = __builtin_amdgcn_wmma_f32_16x16x4_f32(
        false, a1, false, b1, (short)0, tacc, false, false);

    float* p = actBase + (size_t)i * OUT_DIM;
#pragma unroll
    for (int r = 0; r < 8; ++r)
      __builtin_nontemporal_store(d[r], p + (size_t)r * IN_DIM * OUT_DIM);

    acc += d;
  }

  float* py = y + (size_t)(bbase + rowOff) * OUT_DIM + obase + nlo;
#pragma unroll
  for (int r = 0; r < 8; ++r)
    py[(size_t)r * OUT_DIM] = acc[r];
}

// ---------------------------------------------------------------------------
extern "C" void kernel_launch(void* const* d_in, const int* in_sizes, int n_in,
                              void* d_out, int out_size, void* d_ws, size_t ws_size,
                              hipStream_t stream) {
  const float* x      = (const float*)d_in[0];  // (4096, 256)
  const float* grids  = (const float*)d_in[1];  // (256, 9)
  const float* coeffs = (const float*)d_in[2];  // (256, 256, 5)

  float* out  = (float*)d_out;
  float* yOut = out;                                  // (4096, 256)
  float* acts = out + (size_t)BATCH * OUT_DIM;        // (4096, 256, 256)

  float* basisPad = (float*)d_ws;                                   // 32 MB
  float* coeffPad = basisPad + (size_t)BATCH * IN_DIM * KPAD;       // +2 MB

  kan_basis_kernel<<<(BATCH * IN_DIM) / 256, 256, 0, stream>>>(x, grids, basisPad);
  kan_repack_kernel<<<(IN_DIM * OUT_DIM) / 256, 256, 0, stream>>>(coeffs, coeffPad);

  dim3 grid(OUT_DIM / 128, BATCH / 16);   // (2, 256): 8 waves x 2 = 16 o-tiles
  kan_acts_wmma_kernel<<<grid, 256, 0, stream>>>(basisPad, coeffPad, yOut, acts);
}